// GraphSpace_68865505624665
// MI455X (gfx1250) — compile-verified
//
#include <hip/hip_runtime.h>

typedef __attribute__((ext_vector_type(2))) float v2f;
typedef __attribute__((ext_vector_type(8))) float v8f;

#define HIDDEN 64   // input/hidden feature width
#define OUTF   32   // output feature width

// ---------------------------------------------------------------------------
// Degree / normalization
// ---------------------------------------------------------------------------
__global__ void deg_init_kernel(float* __restrict__ deg, int n) {
    int i = blockIdx.x * blockDim.x + threadIdx.x;
    if (i < n) deg[i] = 1.0f;  // self-loop contributes 1 to in-degree
}

__global__ void deg_accum_kernel(const int* __restrict__ dst, float* __restrict__ deg, int e) {
    int i = blockIdx.x * blockDim.x + threadIdx.x;
    if (i < e) atomicAdd(&deg[dst[i]], 1.0f);
}

__global__ void deg_rsqrt_kernel(float* __restrict__ deg, int n) {
    int i = blockIdx.x * blockDim.x + threadIdx.x;
    if (i < n) {
        float d = deg[i];
        deg[i] = (d > 0.0f) ? rsqrtf(d) : 0.0f;  // d >= 1 always, but match ref
    }
}

// ---------------------------------------------------------------------------
// GEMM  H[N x FOUT] = A[N x 64] * W[64 x FOUT]  via V_WMMA_F32_16X16X4_F32
// One wave computes a 16-row x FOUT tile. K=64 in steps of 4.
// ---------------------------------------------------------------------------
template <int FOUT>
__global__ __launch_bounds__(128) void gemm_wmma_f32_kernel(
    const float* __restrict__ A, const float* __restrict__ W,
    float* __restrict__ H, int nRows) {
    constexpr int NT = FOUT / 16;
    const int nTiles = (nRows + 15) >> 4;
    const int wave = blockIdx.x * (blockDim.x >> 5) + (threadIdx.x >> 5);
    if (wave >= nTiles) return;               // wave-uniform branch: EXEC stays full

    const int lane  = threadIdx.x & 31;
    const int lrow  = lane & 15;              // M (for A) / N (for B, D)
    const int khalf = (lane >> 4) * 2;        // 0 or 2: which K-pair this half-wave owns

    int arow = wave * 16 + lrow;
    int arowc = arow < nRows ? arow : (nRows - 1);      // clamp loads at the tail
    const float* __restrict__ Ap = A + (size_t)arowc * HIDDEN;

    v8f acc[NT];
#pragma unroll
    for (int t = 0; t < NT; ++t) acc[t] = (v8f){0.f,0.f,0.f,0.f,0.f,0.f,0.f,0.f};

#pragma unroll
    for (int kk = 0; kk < HIDDEN; kk += 4) {
        v2f a;
        a.x = Ap[kk + khalf];
        a.y = Ap[kk + khalf + 1];
#pragma unroll
        for (int t = 0; t < NT; ++t) {
            const int col = t * 16 + lrow;
            v2f b;
            b.x = W[(size_t)(kk + khalf)     * FOUT + col];
            b.y = W[(size_t)(kk + khalf + 1) * FOUT + col];
            acc[t] = __builtin_amdgcn_wmma_f32_16x16x4_f32(
                false, a, false, b, (short)0, acc[t], false, false);
        }
    }

    // D layout: VGPR r -> M = r + (lane>=16 ? 8 : 0), N = lane&15
    const int mofs = (lane >> 4) * 8;
#pragma unroll
    for (int t = 0; t < NT; ++t) {
#pragma unroll
        for (int r = 0; r < 8; ++r) {
            int row = wave * 16 + mofs + r;
            if (row < nRows)
                H[(size_t)row * FOUT + t * 16 + lrow] = acc[t][r];
        }
    }
}

// ---------------------------------------------------------------------------
// Aggregation-buffer init: agg[i][f] = bias[f] + h[i][f] * dinv[i]^2  (self loop)
// ---------------------------------------------------------------------------
template <int F>
__global__ void agg_init_kernel(const float* __restrict__ h, const float* __restrict__ dinv,
                                const float* __restrict__ bias, float* __restrict__ agg, int n) {
    long long idx = (long long)blockIdx.x * blockDim.x + threadIdx.x;
    long long total = (long long)n * F;
    if (idx < total) {
        int node = (int)(idx / F);
        int f    = (int)(idx - (long long)node * F);
        float dv = dinv[node];
        agg[idx] = bias[f] + h[idx] * dv * dv;
    }
}

// ---------------------------------------------------------------------------
// Edge scatter: one edge per wave32. agg[dst] += h[src] * dinv[src]*dinv[dst]
// F=64: each lane handles features {lane, lane+32}; F=32: feature = lane.
// ---------------------------------------------------------------------------
template <int F>
__global__ void scatter_kernel(const int* __restrict__ srcIdx, const int* __restrict__ dstIdx,
                               const float* __restrict__ dinv, const float* __restrict__ h,
                               float* __restrict__ agg, int e) {
    int wid = blockIdx.x * (blockDim.x >> 5) + (threadIdx.x >> 5);
    if (wid >= e) return;
    int lane = threadIdx.x & 31;
    int s = srcIdx[wid];
    int d = dstIdx[wid];
    float w = dinv[s] * dinv[d];
    const float* __restrict__ hs = h + (size_t)s * F;
    float* __restrict__ ad = agg + (size_t)d * F;
    atomicAdd(&ad[lane], hs[lane] * w);
    if (F == 64) atomicAdd(&ad[lane + 32], hs[lane + 32] * w);
}

// ---------------------------------------------------------------------------
// Host-side orchestration
// ---------------------------------------------------------------------------
extern "C" void kernel_launch(void* const* d_in, const int* in_sizes, int n_in,
                              void* d_out, int out_size, void* d_ws, size_t ws_size,
                              hipStream_t stream) {
    const float* x   = (const float*)d_in[0];     // [N, 64]
    const int*   ei  = (const int*)d_in[1];       // [2, E] (canonicalized int32)
    const float* W1  = (const float*)d_in[2];     // [64, 64]
    const float* b1  = (const float*)d_in[3];     // [64]
    const float* W2  = (const float*)d_in[4];     // [64, 32]
    const float* b2  = (const float*)d_in[5];     // [32]
    float*       out = (float*)d_out;             // [N, 32]

    const int N = in_sizes[0] / HIDDEN;
    const int E = in_sizes[1] / 2;
    const int* src = ei;                          // edge_index[0]
    const int* dst = ei + E;                      // edge_index[1]

    // Workspace layout (floats): dinv[N] | h1[N*64] | agg1[N*64] | h2[N*32]
    float* dinv = (float*)d_ws;
    float* h1   = dinv + N;
    float* agg1 = h1 + (size_t)N * HIDDEN;
    float* h2   = agg1 + (size_t)N * HIDDEN;

    const int TB = 256;
    const int wavesPerBlk = TB / 32;

    // --- normalization ---
    deg_init_kernel<<<(N + TB - 1) / TB, TB, 0, stream>>>(dinv, N);
    deg_accum_kernel<<<(E + TB - 1) / TB, TB, 0, stream>>>(dst, dinv, E);
    deg_rsqrt_kernel<<<(N + TB - 1) / TB, TB, 0, stream>>>(dinv, N);

    // --- layer 1: h1 = x @ W1 ; agg1 = b1 + self + scatter ---
    {
        int nTiles = (N + 15) / 16;
        int blocks = (nTiles + 3) / 4;            // 4 waves / 128-thread block
        gemm_wmma_f32_kernel<HIDDEN><<<blocks, 128, 0, stream>>>(x, W1, h1, N);
    }
    {
        long long total = (long long)N * HIDDEN;
        agg_init_kernel<HIDDEN><<<(int)((total + TB - 1) / TB), TB, 0, stream>>>(h1, dinv, b1, agg1, N);
        scatter_kernel<HIDDEN><<<(E + wavesPerBlk - 1) / wavesPerBlk, TB, 0, stream>>>(src, dst, dinv, h1, agg1, E);
    }

    // --- layer 2: h2 = agg1 @ W2 ; out = b2 + self + scatter ---
    {
        int nTiles = (N + 15) / 16;
        int blocks = (nTiles + 3) / 4;
        gemm_wmma_f32_kernel<OUTF><<<blocks, 128, 0, stream>>>(agg1, W2, h2, N);
    }
    {
        long long total = (long long)N * OUTF;
        agg_init_kernel<OUTF><<<(int)((total + TB - 1) / TB), TB, 0, stream>>>(h2, dinv, b2, out, N);
        scatter_kernel<OUTF><<<(E + wavesPerBlk - 1) / wavesPerBlk, TB, 0, stream>>>(src, dst, dinv, h2, out, E);
    }
}